// DensityExtractor_9105330667608
// MI455X (gfx1250) — compile-verified
//
#include <hip/hip_runtime.h>

// CDNA5 / gfx1250: wave32, WMMA f32 16x16x4 used for block-parallel exclusive
// cumsum (log-domain cumprod) in the NeRF volume-rendering weight kernel.

typedef __attribute__((ext_vector_type(2))) float v2f;
typedef __attribute__((ext_vector_type(8))) float v8f;

#define NSAMP 128
#define WAVE_RAYS 16
#define ROWP 130                      // padded LDS row stride (floats)
#define LT_FLOATS (WAVE_RAYS * ROWP)  // 2080
#define CS_OFF  LT_FLOATS             // 16 rays x 8 chunk sums
#define CAR_OFF (LT_FLOATS + 128)     // 16 rays x 8 exclusive carries
#define WAVE_FLOATS (LT_FLOATS + 256) // 2336 floats per wave

__global__ __launch_bounds__(256) void nerf_weights_kernel(
    const float* __restrict__ rf,     // [N,128,4]
    const float* __restrict__ depth,  // [N,128]
    const float* __restrict__ rdir,   // [N,3]
    float* __restrict__ out,          // [N,128]
    int N)
{
  extern __shared__ float smem[];
  const int lane = threadIdx.x & 31;
  const int wave = threadIdx.x >> 5;
  float* LT  = smem + wave * WAVE_FLOATS;  // log(1-alpha+1e-10), later weights
  float* CS  = LT + CS_OFF;                // per-chunk sums
  float* CAR = LT + CAR_OFF;               // exclusive inter-chunk carries

  const int rayBase = (blockIdx.x * 8 + wave) * WAVE_RAYS;
  const int hi = lane >> 4;   // half-wave select (0/1)
  const int lo = lane & 15;

  // per-ray direction norms (lanes 0..15 hold rays 0..15 of this wave's tile)
  int rr = min(rayBase + lo, N - 1);
  float dx = rdir[rr * 3 + 0], dy = rdir[rr * 3 + 1], dz = rdir[rr * 3 + 2];
  float dnv = sqrtf(dx * dx + dy * dy + dz * dz);

  const float4* rf4 = (const float4*)rf;

  // ---- Pass A: coalesced stream, compute log-transmittance per sample ----
  for (int it = 0; it < 64; ++it) {
    int rl  = it >> 2;                        // ray-local 0..15
    int ray = min(rayBase + rl, N - 1);
    int s   = ((it & 3) << 5) + lane;         // sample 0..127
    float dn = __shfl(dnv, rl, 32);
    long idx = (long)ray * NSAMP + s;
    float4 q = rf4[idx];                      // 16B/lane, 512B/wave coalesced
    float sigma = fmaxf(q.w, 0.f);
    float d0 = depth[idx];
    float d1 = depth[(long)ray * NSAMP + min(s + 1, NSAMP - 1)];
    float dist = (s == NSAMP - 1) ? 1e10f : (d1 - d0);
    float x = sigma * dist * dn;
    float e = __expf(-x);                     // 1 - alpha
    float lt = __logf(e + 1e-10f);            // log(1 - alpha + 1e-10)
    LT[rl * ROWP + s] = lt;
    // 16-wide half-wave reduction -> per-chunk sums (chunks of 16 samples)
    float sum = lt;
    sum += __shfl_xor(sum, 1, 32);
    sum += __shfl_xor(sum, 2, 32);
    sum += __shfl_xor(sum, 4, 32);
    sum += __shfl_xor(sum, 8, 32);
    if (lo == 0) CS[rl * 8 + ((it & 3) << 1) + hi] = sum;
  }
  __syncthreads();

  // ---- Pass B: exclusive scan of 8 chunk sums per ray ----
  if (lane < 16) {
    float c = 0.f;
    for (int k = 0; k < 8; ++k) {
      CAR[lane * 8 + k] = c;
      c += CS[lane * 8 + k];
    }
  }
  __syncthreads();

  // ---- Pass C: per-chunk exclusive cumsum via chained WMMA f32 16x16x4 ----
  // D(16 rays x 16 samples) = X(16x16) * U_strict_upper + carry
  for (int c = 0; c < 8; ++c) {
    v8f acc;
#pragma unroll
    for (int v = 0; v < 8; ++v)
      acc[v] = CAR[(v + 8 * hi) * 8 + c];     // C[r,j] = carry_r (broadcast)
#pragma unroll
    for (int kb = 0; kb < 4; ++kb) {
      // A layout (16x4 f32): lane -> M = lane%16; VGPR0 = K 0|2, VGPR1 = K 1|3
      int base = 16 * c + 4 * kb + 2 * hi;
      v2f a = *(const v2f*)&LT[lo * ROWP + base];
      // B layout (4x16 f32): VGPR v, col = lane%16, row k = v + 2*(lane/16)
      int k0 = 4 * kb + 2 * hi;
      v2f b;
      b.x = (k0 < lo) ? 1.f : 0.f;            // strictly-upper ones => exclusive
      b.y = (k0 + 1 < lo) ? 1.f : 0.f;
      acc = __builtin_amdgcn_wmma_f32_16x16x4_f32(
          /*neg_a=*/false, a, /*neg_b=*/false, b,
          /*c_mod=*/(short)0, acc, /*reuse_a=*/false, /*reuse_b=*/false);
    }
    // weights = alpha * exp(exclusive cumsum); overwrite this chunk in LDS
#pragma unroll
    for (int v = 0; v < 8; ++v) {
      int rowD = v + 8 * hi;                  // D: VGPR v -> rows v / v+8
      int colD = 16 * c + lo;
      float lt = LT[rowD * ROWP + colD];
      float alpha = 1.0f + 1e-10f - __expf(lt);
      LT[rowD * ROWP + colD] = alpha * __expf(acc[v]);
    }
  }
  __syncthreads();

  // ---- Pass D: coalesced stores ----
  for (int it = 0; it < 64; ++it) {
    int rl  = it >> 2;
    int ray = min(rayBase + rl, N - 1);
    int s   = ((it & 3) << 5) + lane;
    out[(long)ray * NSAMP + s] = LT[rl * ROWP + s];
  }
}

extern "C" void kernel_launch(void* const* d_in, const int* in_sizes, int n_in,
                              void* d_out, int out_size, void* d_ws, size_t ws_size,
                              hipStream_t stream) {
  const float* rf    = (const float*)d_in[0];   // radiance_field [N,128,4]
  const float* depth = (const float*)d_in[1];   // depth_values   [N,128]
  const float* rdir  = (const float*)d_in[2];   // ray_directions [N,3]
  float* out = (float*)d_out;

  int N = in_sizes[1] / NSAMP;                  // 65536
  int blocks = (N + 8 * WAVE_RAYS - 1) / (8 * WAVE_RAYS);   // 128 rays/block
  size_t shmem = 8 * WAVE_FLOATS * sizeof(float);           // ~74.75 KB

  nerf_weights_kernel<<<blocks, 256, shmem, stream>>>(rf, depth, rdir, out, N);
}